// GINConv_no_nn_multi_5239860101132
// MI455X (gfx1250) — compile-verified
//
#include <hip/hip_runtime.h>

// GIN (no MLP) x3 layers with concat, eps=0, scale=1.
// x: [N,128] f32, edge_index: [2,E] i64. out: [N, 512] f32 where
// out[:, l*128:(l+1)*128] = x_l, x_l = segment_sum(x_{l-1}[src], dst) + x_{l-1}.
//
// MI455X strategy: build CSR once in workspace (3.4MB); each layer is an
// atomic-free gather-sum whose working set (two 51MB slices + CSR) is resident
// in the 192MB L2. One wave32 per node; 512B coalesced b128 row loads;
// shfl-broadcast neighbor ids; global_prefetch_b8 to hide L2 gather latency;
// NT loads for stream-once data; NT stores for the final (never re-read) slice.

#define D_FEAT   128
#define ROW_STR  512    // 4 * D_FEAT floats per output row
#define LAYERS   3

typedef float v4f __attribute__((ext_vector_type(4)));

__global__ void k_zero(int* __restrict__ p, int n) {
  int i = blockIdx.x * blockDim.x + threadIdx.x;
  if (i < n) p[i] = 0;
}

__global__ void k_count(const long long* __restrict__ dst, int E,
                        int* __restrict__ counts) {
  int e = blockIdx.x * blockDim.x + threadIdx.x;
  if (e < E) {
    long long d = __builtin_nontemporal_load(dst + e);   // consumed once: NT
    atomicAdd(&counts[(int)d], 1);
  }
}

// Single-workgroup exclusive scan over counts[0..n-1]; writes row_ptr in place
// (counts becomes row_ptr), duplicates into cursor, appends total at [n].
__global__ __launch_bounds__(1024) void k_scan(int* __restrict__ counts,
                                               int* __restrict__ cursor, int n) {
  __shared__ int part[1024];
  const int t = threadIdx.x;
  const int chunk = (n + 1023) / 1024;
  const int b = t * chunk;
  const int e = (b + chunk < n) ? (b + chunk) : n;
  int s = 0;
  for (int i = b; i < e; ++i) s += counts[i];
  part[t] = s;
  __syncthreads();
  for (int off = 1; off < 1024; off <<= 1) {   // Hillis–Steele inclusive scan
    int v = part[t];
    int add = (t >= off) ? part[t - off] : 0;
    __syncthreads();
    part[t] = v + add;
    __syncthreads();
  }
  int running = (t == 0) ? 0 : part[t - 1];    // exclusive base for this chunk
  for (int i = b; i < e; ++i) {
    int c = counts[i];
    counts[i] = running;
    cursor[i] = running;
    running += c;
  }
  if (t == 1023) counts[n] = running;          // total edge count
}

__global__ void k_fill(const long long* __restrict__ src,
                       const long long* __restrict__ dst, int E,
                       int* __restrict__ cursor, int* __restrict__ col) {
  int e = blockIdx.x * blockDim.x + threadIdx.x;
  if (e < E) {
    long long d = __builtin_nontemporal_load(dst + e);
    long long s = __builtin_nontemporal_load(src + e);
    int p = atomicAdd(&cursor[(int)d], 1);
    col[p] = (int)s;
  }
}

// out[:, 0:128] = x  (layer-0 slice). x is read exactly once -> NT load.
__global__ void k_copy0(const v4f* __restrict__ x, float* __restrict__ out,
                        int n_nodes) {
  int i = blockIdx.x * blockDim.x + threadIdx.x;
  int node = i >> 5, q = i & 31;
  if (node < n_nodes) {
    v4f v = __builtin_nontemporal_load(x + (size_t)node * 32 + q);
    *(v4f*)(out + (size_t)node * ROW_STR + q * 4) = v;  // re-read by layer 1: RT
  }
}

// One wave32 per node: acc = prev[node]; acc += prev[nbr] for all CSR nbrs.
// prev/next point at column slices of d_out (row stride ROW_STR).
template <bool LAST>
__global__ __launch_bounds__(256) void k_gin(const float* __restrict__ prev,
                                             float* __restrict__ next,
                                             const int* __restrict__ rowptr,
                                             const int* __restrict__ col,
                                             int n_nodes) {
  const int lane = threadIdx.x & 31;
  const int node = blockIdx.x * 8 + (threadIdx.x >> 5);
  if (node >= n_nodes) return;
  const size_t rb = (size_t)node * ROW_STR + lane * 4;
  v4f acc = *(const v4f*)(prev + rb);                   // (1+eps)*x_i, eps=0
  const int s = rowptr[node], e = rowptr[node + 1];
  for (int base = s; base < e; base += 32) {
    const int rem = e - base;
    const int cnt = rem < 32 ? rem : 32;
    int my = (lane < cnt) ? col[base + lane] : 0;       // coalesced id batch
    if (lane < cnt) {                                   // warm caches: 512B row
      const float* r = prev + (size_t)my * ROW_STR;     // = 4x128B lines
      __builtin_prefetch(r, 0, 3);                      // global_prefetch_b8
      __builtin_prefetch(r + 64, 0, 3);                 // +256B
    }
    int k = 0;
    for (; k + 4 <= cnt; k += 4) {                      // 4 loads in flight
      const int n0 = __shfl(my, k + 0, 32);
      const int n1 = __shfl(my, k + 1, 32);
      const int n2 = __shfl(my, k + 2, 32);
      const int n3 = __shfl(my, k + 3, 32);
      const v4f v0 = *(const v4f*)(prev + (size_t)n0 * ROW_STR + lane * 4);
      const v4f v1 = *(const v4f*)(prev + (size_t)n1 * ROW_STR + lane * 4);
      const v4f v2 = *(const v4f*)(prev + (size_t)n2 * ROW_STR + lane * 4);
      const v4f v3 = *(const v4f*)(prev + (size_t)n3 * ROW_STR + lane * 4);
      acc += v0 + v1 + v2 + v3;
    }
    for (; k < cnt; ++k) {
      const int nbr = __shfl(my, k, 32);
      acc += *(const v4f*)(prev + (size_t)nbr * ROW_STR + lane * 4);
    }
  }
  if (LAST)                                             // never re-read: stream
    __builtin_nontemporal_store(acc, (v4f*)(next + rb));
  else
    *(v4f*)(next + rb) = acc;                           // next layer's input: RT
}

// ---- fallback path (workspace too small): atomic scatter-add ----
__global__ void k_copy_slice(const float* __restrict__ prev,
                             float* __restrict__ next, int n_nodes) {
  int i = blockIdx.x * blockDim.x + threadIdx.x;
  int node = i >> 5, q = i & 31;
  if (node < n_nodes)
    *(v4f*)(next + (size_t)node * ROW_STR + q * 4) =
        *(const v4f*)(prev + (size_t)node * ROW_STR + q * 4);
}

__global__ __launch_bounds__(256) void k_scatter(const long long* __restrict__ src,
                                                 const long long* __restrict__ dst,
                                                 const float* __restrict__ prev,
                                                 float* __restrict__ next, int E) {
  const int lane = threadIdx.x & 31;
  const int e = blockIdx.x * 8 + (threadIdx.x >> 5);    // one wave per edge
  if (e >= E) return;
  const int s = (int)src[e], d = (int)dst[e];
  const v4f v = *(const v4f*)(prev + (size_t)s * ROW_STR + lane * 4);
  float* p = next + (size_t)d * ROW_STR + lane * 4;
  atomicAdd(p + 0, v.x); atomicAdd(p + 1, v.y);
  atomicAdd(p + 2, v.z); atomicAdd(p + 3, v.w);
}

extern "C" void kernel_launch(void* const* d_in, const int* in_sizes, int n_in,
                              void* d_out, int out_size, void* d_ws, size_t ws_size,
                              hipStream_t stream) {
  const float*     x   = (const float*)d_in[0];
  const long long* ei  = (const long long*)d_in[1];
  const int n_nodes = in_sizes[0] / D_FEAT;
  const int E       = in_sizes[1] / 2;
  const long long* src = ei;         // row 0: message sources
  const long long* dst = ei + E;     // row 1: aggregation targets
  float* out = (float*)d_out;

  {  // layer-0 slice
    const int tot = n_nodes * 32;
    k_copy0<<<(tot + 255) / 256, 256, 0, stream>>>((const v4f*)x, out, n_nodes);
  }

  const size_t need = (size_t)(n_nodes + 1) * 4 + (size_t)n_nodes * 4 + (size_t)E * 4;
  if (ws_size >= need) {
    int* rowptr = (int*)d_ws;                 // n_nodes + 1
    int* cursor = rowptr + (n_nodes + 1);     // n_nodes
    int* col    = cursor + n_nodes;           // E
    k_zero <<<(n_nodes + 1 + 255) / 256, 256, 0, stream>>>(rowptr, n_nodes + 1);
    k_count<<<(E + 255) / 256,           256, 0, stream>>>(dst, E, rowptr);
    k_scan <<<1, 1024, 0, stream>>>(rowptr, cursor, n_nodes);
    k_fill <<<(E + 255) / 256,           256, 0, stream>>>(src, dst, E, cursor, col);
    for (int l = 1; l <= LAYERS; ++l) {
      const float* prev = out + (size_t)(l - 1) * D_FEAT;
      float*       next = out + (size_t)l * D_FEAT;
      if (l == LAYERS)
        k_gin<true><<<(n_nodes + 7) / 8, 256, 0, stream>>>(prev, next, rowptr, col, n_nodes);
      else
        k_gin<false><<<(n_nodes + 7) / 8, 256, 0, stream>>>(prev, next, rowptr, col, n_nodes);
    }
  } else {
    for (int l = 1; l <= LAYERS; ++l) {
      const float* prev = out + (size_t)(l - 1) * D_FEAT;
      float*       next = out + (size_t)l * D_FEAT;
      const int tot = n_nodes * 32;
      k_copy_slice<<<(tot + 255) / 256, 256, 0, stream>>>(prev, next, n_nodes);
      k_scatter  <<<(E + 7) / 8,        256, 0, stream>>>(src, dst, prev, next, E);
    }
  }
}